// DynamicRouting_60447369724743
// MI455X (gfx1250) — compile-verified
//
#include <hip/hip_runtime.h>
#include <math.h>

typedef __attribute__((ext_vector_type(2))) float v2f;
typedef __attribute__((ext_vector_type(8))) float v8f;

#define NG    8      // groups
#define FI    64
#define FO    64
#define HW    4096   // 64*64 pixels per image
#define NB    16     // batch
#define PSTR  514    // LDS floats per pixel (512 + 2 pad -> conflict-free banks)
#define WPG   4      // waves per workgroup
#define TP    16     // pixels per wave tile

__device__ __forceinline__ float sigmoidf_(float x) {
    return 1.0f / (1.0f + __expf(-x));
}

extern "C" __global__ __launch_bounds__(128)
void dynrout_fused_kernel(const float* __restrict__ x,
                          const float* __restrict__ w,
                          const float* __restrict__ bias,
                          float* __restrict__ out)
{
    extern __shared__ float lds[];

    const int tid  = threadIdx.x;
    const int lane = tid & 31;
    const int wv   = tid >> 5;
    const int l15  = lane & 15;     // pixel-in-tile (A row / D col role)
    const int hf   = lane >> 4;     // half-wave id (K split / f-half)

    float* my = lds + wv * (TP * PSTR);   // wave-private con buffer: [16 pixels][514]

    const int wid = blockIdx.x * WPG + wv;       // 0..4095 wave tiles
    const int b   = wid >> 8;                    // 256 tiles per image
    const int p0  = (wid & 255) * TP;            // first pixel of tile

    const float* xb = x + (size_t)b * (NG * FI) * HW;

    // ---------------- conv phase: con[p][g*64+f] via V_WMMA_F32_16X16X4_F32 ----------------
    for (int g = 0; g < NG; ++g) {
        v8f acc[4];
        #pragma unroll
        for (int t = 0; t < 4; ++t) acc[t] = (v8f){};

        const float* xg = xb + (size_t)(g * FI) * HW + p0 + l15;
        const float* wg = w + (size_t)(g * FO) * FI;

        #pragma unroll
        for (int k0 = 0; k0 < FI; k0 += 4) {
            const int fi0 = k0 + 2 * hf;            // this half-wave's K rows
            v2f A;
            A.x = xg[(size_t)fi0 * HW];             // coalesced: 16 consecutive pixels
            A.y = xg[(size_t)(fi0 + 1) * HW];

            #pragma unroll
            for (int t = 0; t < 4; ++t) {
                // B (K=4 x N=16): lane holds column fo = t*16 + l15, K pair (fi0, fi0+1)
                const float* bp = wg + (size_t)(t * 16 + l15) * FI + fi0;
                v2f Bv = *(const v2f*)bp;           // 8B aligned (fi0 even)
                acc[t] = __builtin_amdgcn_wmma_f32_16x16x4_f32(
                    false, A, false, Bv, (short)0, acc[t], false, false);
            }
        }

        // store D tiles: VGPR r, lane -> (pixel m = r + 8*hf, f = t*16 + l15)
        #pragma unroll
        for (int t = 0; t < 4; ++t) {
            #pragma unroll
            for (int r = 0; r < 8; ++r) {
                const int m = r + 8 * hf;
                my[m * PSTR + g * FO + t * 16 + l15] = acc[t][r];
            }
        }
    }

    // ---------------- routing phase (wave-private, no barrier) ----------------
    // lane owns pixel p = l15, f-half hf (f in [hf*32, hf*32+32))
    const float* pb = my + l15 * PSTR + hf * 32;

    // Gram is symmetric: accumulate only upper triangle (36 pairs)
    float gram[NG][NG];
    #pragma unroll
    for (int g = 0; g < NG; ++g)
        #pragma unroll
        for (int g2 = g; g2 < NG; ++g2) gram[g][g2] = 0.0f;

    for (int j = 0; j < 32; j += 2) {
        v2f c[NG];
        #pragma unroll
        for (int g = 0; g < NG; ++g)
            c[g] = *(const v2f*)(pb + g * FO + j);      // ds_load_b64
        #pragma unroll
        for (int g = 0; g < NG; ++g)
            #pragma unroll
            for (int g2 = g; g2 < NG; ++g2) {
                gram[g][g2] = fmaf(c[g].x, c[g2].x, gram[g][g2]);
                gram[g][g2] = fmaf(c[g].y, c[g2].y, gram[g][g2]);
            }
    }
    // combine the two f-halves of this pixel (lanes l and l^16)
    #pragma unroll
    for (int g = 0; g < NG; ++g)
        #pragma unroll
        for (int g2 = g; g2 < NG; ++g2)
            gram[g][g2] += __shfl_xor(gram[g][g2], 16, 32);

    // symmetric accessor (resolves at compile time in unrolled loops)
    #define GR(a, b) ((a) <= (b) ? gram[(a)][(b)] : gram[(b)][(a)])

    // iter 0: alpha = 0.5 -> beta1 = 0.5 * row-sum(Gram)
    float beta[NG], alpha[NG];
    #pragma unroll
    for (int g = 0; g < NG; ++g) {
        float s = 0.0f;
        #pragma unroll
        for (int g2 = 0; g2 < NG; ++g2) s += GR(g, g2);
        beta[g] = 0.5f * s;
    }
    // iter 1: beta2 = beta1 + Gram @ sigmoid(beta1)
    #pragma unroll
    for (int g = 0; g < NG; ++g) alpha[g] = sigmoidf_(beta[g]);
    #pragma unroll
    for (int g = 0; g < NG; ++g) {
        float s = 0.0f;
        #pragma unroll
        for (int g2 = 0; g2 < NG; ++g2) s = fmaf(alpha[g2], GR(g, g2), s);
        beta[g] += s;
    }
    // iter 2 weights
    #pragma unroll
    for (int g = 0; g < NG; ++g) alpha[g] = sigmoidf_(beta[g]);
    #undef GR

    // out[b][f][p] = sum_g alpha2[g] * con[p][g*64+f] + bias[f]
    float* op = out + (size_t)b * FO * HW + p0 + l15;
    for (int j = 0; j < 32; j += 2) {
        const int f = hf * 32 + j;
        v2f bv = *(const v2f*)(bias + f);               // 8B aligned (f even)
        float v0 = bv.x, v1 = bv.y;
        #pragma unroll
        for (int g = 0; g < NG; ++g) {
            v2f c = *(const v2f*)(pb + g * FO + j);     // ds_load_b64
            v0 = fmaf(alpha[g], c.x, v0);
            v1 = fmaf(alpha[g], c.y, v1);
        }
        op[(size_t)f * HW]       = v0;   // coalesced: 16 consecutive pixels per half-wave
        op[(size_t)(f + 1) * HW] = v1;
    }
}

extern "C" void kernel_launch(void* const* d_in, const int* in_sizes, int n_in,
                              void* d_out, int out_size, void* d_ws, size_t ws_size,
                              hipStream_t stream) {
    const float* x    = (const float*)d_in[0];
    const float* w    = (const float*)d_in[1];
    const float* bias = (const float*)d_in[2];
    float* out = (float*)d_out;

    const int total_tiles = NB * (HW / TP);          // 4096 wave tiles
    const int blocks = total_tiles / WPG;            // 1024
    const size_t shmem = (size_t)WPG * TP * PSTR * sizeof(float);  // ~128.5 KB

    dynrout_fused_kernel<<<blocks, 128, shmem, stream>>>(x, w, bias, out);
}